// ExtractTensorPatches_57123065037114
// MI455X (gfx1250) — compile-verified
//
#include <hip/hip_runtime.h>
#include <hip/hip_bf16.h>
#include <stdint.h>

// ExtractTensorPatches: x(16,3,512,512) f32, window 16x16, stride 8x8
// -> out(16, 63*63, 3, 16, 16) f32.
//
// Pure data movement (0 FLOPs). HBM floor ~245MB @ 23.3TB/s ~= 10.5us.
// Per-(b,c,ho) block stages 16 input rows into LDS with the CDNA5 async
// global->LDS engine (ASYNCcnt-tracked), then streams all 63 sliding patches
// out with coalesced non-temporal b128 stores. LDS rows padded to 528 floats
// (528 % 64 == 16 banks) so the b128 patch reads are bank-conflict-free per
// 16-lane phase.

typedef __attribute__((ext_vector_type(4))) float v4f;

#define B_   16
#define C_   3
#define H_   512
#define W_   512
#define HO_  63
#define WO_  63
#define P_   (HO_ * WO_)      // 3969 patches
#define LDSW 528              // padded LDS row stride in floats

__global__ __launch_bounds__(256)
void ExtractTensorPatches_57123065037114_kernel(const float* __restrict__ x,
                                                float* __restrict__ out) {
    __shared__ __align__(16) float smem[16 * LDSW];   // 33792 B

    const int ho  = blockIdx.x;   // 0..62
    const int c   = blockIdx.y;   // 0..2
    const int b   = blockIdx.z;   // 0..15
    const int tid = threadIdx.x;  // 0..255

    // ---- Phase 1: async-load 16 input rows (16 x 512 floats) into LDS ----
    // idx = k*256 + tid: col4 = idx & 127 is loop-invariant, row = idx >> 7
    // steps by 2 each iteration -> pure pointer-bump loop.
    const float* src = x + (((size_t)(b * C_ + c) * H_ + (size_t)ho * 8) * (size_t)W_);
    {
        const int row0 = tid >> 7;        // 0..1
        const int col4 = tid & 127;       // 0..127
        unsigned lds_addr = (unsigned)(size_t)(void*)smem
                          + (unsigned)(row0 * LDSW + col4 * 4) * 4u;
        uint64_t gaddr = (uint64_t)(const void*)(src + (size_t)row0 * W_ + (size_t)col4 * 4);
        #pragma unroll
        for (int k = 0; k < 8; ++k) {
            asm volatile("global_load_async_to_lds_b128 %0, %1, off"
                         :
                         : "v"(lds_addr), "v"(gaddr)
                         : "memory");
            lds_addr += 2u * LDSW * 4u;   // +2 rows in LDS
            gaddr    += 2ull * W_ * 4ull; // +2 rows in global
        }
    }
    asm volatile("s_wait_asynccnt 0" ::: "memory");
    __syncthreads();

    // ---- Phase 2: emit 63 patches (16x16 floats each) for this (b,c,ho) ----
    // idx = wo*64 + i*4 + v with idx += 256  =>  r = i*4+v is per-thread
    // invariant and wo steps by 4. A 32-lane wave (fixed wo, i=0..7, v=0..3)
    // stores 512B contiguous and reads LDS banks 4*(4i+v) mod 64.
    {
        const int r = tid & 63;           // i*4 + v
        const int i = r >> 2;
        const int v = r & 3;
        const float* sp = &smem[i * LDSW + v * 4];
        float* op = out + (((size_t)(b * P_ + ho * WO_) * C_ + c) * 256) + r * 4;

        const float* spw = sp + (tid >> 6) * 8;                    // + wo0 cols
        float*       opw = op + (size_t)(tid >> 6) * (C_ * 256);   // + wo0 slabs
        for (int wo = tid >> 6; wo < WO_; wo += 4) {
            v4f val = *(const v4f*)spw;
            __builtin_nontemporal_store(val, (v4f*)opw);
            spw += 4 * 8;                  // 4 patches right = 32 floats
            opw += (size_t)4 * C_ * 256;   // 4 output patch slabs
        }
    }
}

extern "C" void kernel_launch(void* const* d_in, const int* in_sizes, int n_in,
                              void* d_out, int out_size, void* d_ws, size_t ws_size,
                              hipStream_t stream) {
    const float* x = (const float*)d_in[0];
    float* out = (float*)d_out;
    dim3 grid(HO_, C_, B_);   // one block per (ho, c, b)
    ExtractTensorPatches_57123065037114_kernel<<<grid, 256, 0, stream>>>(x, out);
}